// Nms_24043226923896
// MI455X (gfx1250) — compile-verified
//
#include <hip/hip_runtime.h>
#include <hip/hip_bf16.h>
#include <stdint.h>

// NMS (torchvision semantics) for N=8192, IoU thr 0.5, MAX_OUT=300.
// Phase 1: bitonic argsort by score (desc) in LDS.
// Phase 2: pairwise suppression bitmask (upper triangle), column boxes staged
//          into LDS via CDNA5 async-to-LDS loads.
// Phase 3: single-wave greedy scan over the bitmask with early exit at 300.

#define NMS_N       8192
#define NMS_NB      256      // N / 32 mask words per row
#define NMS_MAXOUT  300

// ---------------------------------------------------------------- Phase 1
__global__ __launch_bounds__(1024) void nms_sort_kernel(
    const float* __restrict__ scores, const float* __restrict__ rois,
    float4* __restrict__ sboxes, int* __restrict__ order) {
  __shared__ unsigned long long keys[NMS_N];   // 64 KB LDS
  const int t = threadIdx.x;

  for (int v = t; v < NMS_N; v += 1024) {
    unsigned u = __float_as_uint(scores[v]);   // scores > 0 -> bits monotonic
    keys[v] = ((unsigned long long)u << 32) | (unsigned)v;
  }
  __syncthreads();

  for (int k = 2; k <= NMS_N; k <<= 1) {
    for (int j = k >> 1; j > 0; j >>= 1) {
      for (int p = t; p < NMS_N / 2; p += 1024) {
        int i   = ((p & ~(j - 1)) << 1) | (p & (j - 1));
        int ixj = i | j;
        bool desc = ((i & k) == 0);            // overall descending sort
        unsigned long long a = keys[i], b = keys[ixj];
        if ((a < b) == desc) { keys[i] = b; keys[ixj] = a; }
      }
      __syncthreads();
    }
  }

  const float4* r4 = (const float4*)rois;
  for (int v = t; v < NMS_N; v += 1024) {
    int oi = (int)(unsigned)(keys[v] & 0xffffffffull);
    order[v]  = oi;
    sboxes[v] = r4[oi];                        // gather into score order
  }
}

// ---------------------------------------------------------------- Phase 2
// grid = (256 col-blocks, 64), block = (32, 4): each wave owns one 32-row tile.
__global__ __launch_bounds__(128) void nms_mask_kernel(
    const float4* __restrict__ sboxes, unsigned* __restrict__ mask) {
  __shared__ float4 cbox[32];
  const int tx = threadIdx.x;
  const int ty = threadIdx.y;
  const int cb = blockIdx.x;                       // column block
  const int rb = blockIdx.y * 4 + ty;              // this wave's row block

  // Stage the 32 column boxes into LDS with the CDNA5 async-to-LDS path.
  if (ty == 0) {
    const float4* src = sboxes + ((size_t)cb << 5) + tx;
    unsigned      lds = (unsigned)(uintptr_t)&cbox[tx];     // low 32b = LDS off
    unsigned long long ga = (unsigned long long)(uintptr_t)src;
    asm volatile("global_load_async_to_lds_b128 %0, %1, off"
                 :: "v"(lds), "v"(ga) : "memory");
    asm volatile("s_wait_asynccnt 0x0" ::: "memory");
  }
  __syncthreads();

  if (cb < rb) return;                             // lower triangle unused

  const int i = (rb << 5) + tx;                    // my row (score order)
  const float4 b = sboxes[i];
  const float area_i = (b.z - b.x) * (b.w - b.y);

  unsigned bits = 0u;
#pragma unroll 8
  for (int kk = 0; kk < 32; ++kk) {
    int j = (cb << 5) + kk;
    float4 c = cbox[kk];
    float xx1 = fmaxf(b.x, c.x), yy1 = fmaxf(b.y, c.y);
    float xx2 = fminf(b.z, c.z), yy2 = fminf(b.w, c.w);
    float w = fmaxf(xx2 - xx1, 0.0f);
    float h = fmaxf(yy2 - yy1, 0.0f);
    float inter  = w * h;
    float area_j = (c.z - c.x) * (c.w - c.y);
    // iou > 0.5  <=>  inter > 0.5*(ai+aj-inter)  <=>  3*inter > ai+aj
    bool sup = (j > i) && (3.0f * inter > area_i + area_j);
    bits |= sup ? (1u << kk) : 0u;
  }
  mask[(size_t)i * NMS_NB + cb] = bits;
}

// ---------------------------------------------------------------- Phase 3
// One wave32. Removal bitmask (256 words = 8192 bits) lives in LDS.
__global__ __launch_bounds__(32) void nms_reduce_kernel(
    const unsigned* __restrict__ mask, const int* __restrict__ order,
    int* __restrict__ out) {
  __shared__ unsigned remS[NMS_NB];
  volatile unsigned* rem = remS;                   // in-order DS ops, 1 wave
  const int lane = threadIdx.x;

#pragma unroll
  for (int s = 0; s < 8; ++s) rem[(s << 5) | lane] = 0u;

  int count = 0;
  for (int i = 0; i < NMS_N; ++i) {
    const int w = i >> 5;
    unsigned wv = rem[w];                          // uniform broadcast read
    if (!((wv >> (i & 31)) & 1u)) {
      if (lane == 0) out[count] = order[i];        // original index
      ++count;
      if (count >= NMS_MAXOUT) break;              // suppression flows forward
      const unsigned* row = mask + (size_t)i * NMS_NB;
#pragma unroll
      for (int s = 0; s < 8; ++s) {
        int wi = (s << 5) | lane;
        if (wi >= w) rem[wi] |= row[wi];           // only written triangle
      }
    }
  }
  for (int v = count + lane; v < NMS_MAXOUT; v += 32) out[v] = -1;
}

// ---------------------------------------------------------------- launch
extern "C" void kernel_launch(void* const* d_in, const int* in_sizes, int n_in,
                              void* d_out, int out_size, void* d_ws, size_t ws_size,
                              hipStream_t stream) {
  const float* rois   = (const float*)d_in[0];   // [8192,4] xyxy
  const float* scores = (const float*)d_in[1];   // [8192]
  int* out = (int*)d_out;                        // [300] int32

  char* ws = (char*)d_ws;
  float4*   sboxes = (float4*)ws;                              // 131072 B
  int*      order  = (int*)(ws + 131072);                      //  32768 B
  unsigned* mask   = (unsigned*)(ws + 131072 + 32768);         // 8 MB

  nms_sort_kernel<<<1, 1024, 0, stream>>>(scores, rois, sboxes, order);
  nms_mask_kernel<<<dim3(NMS_NB, 64), dim3(32, 4), 0, stream>>>(sboxes, mask);
  nms_reduce_kernel<<<1, 32, 0, stream>>>(mask, order, out);
}